// SpatialAdaptiveSynBatchNorm2d_4776003633650
// MI455X (gfx1250) — compile-verified
//
#include <hip/hip_runtime.h>
#include <hip/hip_bf16.h>

// Fixed problem shapes from the reference setup_inputs():
//   x:      (B=8, C=128, H=128, W=128) fp32
//   vector: (B*O=80, K=512)            fp32
//   bbox:   (B=8, O=10, H=128, W=128)  fp32
//   wp_w/bp_w: (C=128, K=512), wp_b/bp_b: (C=128,)
#define BN   8
#define CN   128
#define ON   10
#define HW   16384      // 128*128
#define KN   512
#define MROWS 80        // B*O

typedef float v2f __attribute__((ext_vector_type(2)));
typedef float v8f __attribute__((ext_vector_type(8)));

// ---------------------------------------------------------------------------
// Kernel 1: per-channel batch stats.  One block per channel (128 blocks x 256
// threads).  Reduces 8*16384 = 131072 values -> mean, rstd = rsqrt(var+1e-5).
// stats[0..127] = mean, stats[128..255] = rstd.
// ---------------------------------------------------------------------------
__global__ void bn_stats_kernel(const float* __restrict__ x,
                                float* __restrict__ stats) {
    const int c = blockIdx.x;
    const int t = threadIdx.x;

    float sum = 0.0f, sq = 0.0f;
    for (int b = 0; b < BN; ++b) {
        const float* p = x + (size_t)(b * CN + c) * HW;
        for (int i = t; i < HW; i += 256) {
            float v = p[i];
            sum += v;
            sq = fmaf(v, v, sq);
        }
    }

    __shared__ float ssum[256];
    __shared__ float ssq[256];
    ssum[t] = sum;
    ssq[t]  = sq;
    __syncthreads();
    for (int off = 128; off > 0; off >>= 1) {
        if (t < off) {
            ssum[t] += ssum[t + off];
            ssq[t]  += ssq[t + off];
        }
        __syncthreads();
    }
    if (t == 0) {
        const float invN = 1.0f / (float)(BN * HW);
        float mean = ssum[0] * invN;
        float var  = ssq[0] * invN - mean * mean;
        stats[c]        = mean;
        stats[CN + c]   = rsqrtf(var + 1e-5f);
    }
}

// ---------------------------------------------------------------------------
// Kernel 2: projection GEMMs with full-precision WMMA (V_WMMA_F32_16X16X4_F32).
//   weight = vector @ wp_w.T + wp_b   (80 x 128)
//   bias   = vector @ bp_w.T + bp_b   (80 x 128)
// Grid (8 Ntiles, 5 Mtiles, 2 matrices), one wave32 per 16x16 tile.
// A-frag (16x4 f32): lanes 0-15 hold K={k,k+1}, lanes 16-31 hold K={k+2,k+3},
// per ISA 7.12.2.  B-frag (4x16) mirrors with N striped across lanes.
// D (16x16 f32): VGPR r holds M=r (lanes 0-15) / M=r+8 (lanes 16-31), N=lane&15.
// ---------------------------------------------------------------------------
__global__ __launch_bounds__(32)
void proj_wmma_kernel(const float* __restrict__ vec,
                      const float* __restrict__ wp_w,
                      const float* __restrict__ wp_b,
                      const float* __restrict__ bp_w,
                      const float* __restrict__ bp_b,
                      float* __restrict__ wout,
                      float* __restrict__ bout) {
    const int lane = threadIdx.x;           // 0..31
    const int l    = lane & 15;
    const int hi   = lane >> 4;             // 0 or 1
    const int n0   = blockIdx.x * 16;       // output column tile (c)
    const int m0   = blockIdx.y * 16;       // output row tile (b*o)
    const bool isBias = (blockIdx.z != 0);

    const float* Wm  = isBias ? bp_w : wp_w;
    const float* Bv  = isBias ? bp_b : wp_b;
    float*       Out = isBias ? bout : wout;

    const int arow = m0 + l;                // A row for this lane
    const int bcol = n0 + l;                // B column for this lane

    v8f acc = {0.f, 0.f, 0.f, 0.f, 0.f, 0.f, 0.f, 0.f};

    for (int k = 0; k < KN; k += 4) {
        const int kk = k + 2 * hi;
        v2f a, b;
        a.x = vec[arow * KN + kk];
        a.y = vec[arow * KN + kk + 1];
        b.x = Wm[bcol * KN + kk];           // B[k][n] = W[n][k]
        b.y = Wm[bcol * KN + kk + 1];
        acc = __builtin_amdgcn_wmma_f32_16x16x4_f32(
            /*neg_a=*/false, a, /*neg_b=*/false, b,
            /*c_mod=*/(short)0, acc, /*reuse_a=*/false, /*reuse_b=*/false);
    }

    const float biasv = Bv[bcol];
#pragma unroll
    for (int r = 0; r < 8; ++r) {
        const int m = m0 + r + 8 * hi;
        Out[m * CN + bcol] = acc[r] + biasv;
    }
}

// ---------------------------------------------------------------------------
// Kernel 3: fused bbox-weighted map + batchnorm apply.
// Grid (64 pixel-tiles, 8 batches) x 256 threads; thread = one pixel, loops
// channels x4.  bbox values normalized once per pixel; weight/bias/mean/rstd
// reads are block-uniform -> scalar loads; x load/store fully coalesced.
// ---------------------------------------------------------------------------
__global__ __launch_bounds__(256)
void fused_apply_kernel(const float* __restrict__ x,
                        const float* __restrict__ bbox,
                        const float* __restrict__ stats,
                        const float* __restrict__ wmat,
                        const float* __restrict__ bmat,
                        float* __restrict__ out) {
    const int b   = blockIdx.y;
    const int pix = blockIdx.x * 256 + threadIdx.x;   // 0..16383

    // Per-pixel bbox weights, pre-divided by (sum + eps).
    float bb[ON];
    float s = 0.0f;
#pragma unroll
    for (int o = 0; o < ON; ++o) {
        bb[o] = bbox[(size_t)(b * ON + o) * HW + pix];
        s += bb[o];
    }
    const float inv = 1.0f / (s + 1e-6f);
#pragma unroll
    for (int o = 0; o < ON; ++o) bb[o] *= inv;

    const float* Wt = wmat + b * (ON * CN);   // (10,128), block-uniform
    const float* Bt = bmat + b * (ON * CN);
    const float* meanp = stats;
    const float* rstdp = stats + CN;

    const float* xb = x   + (size_t)b * CN * HW + pix;
    float*       ob = out + (size_t)b * CN * HW + pix;

    for (int c0 = 0; c0 < CN; c0 += 4) {
        float w0 = 0.f, w1 = 0.f, w2 = 0.f, w3 = 0.f;
        float a0 = 0.f, a1 = 0.f, a2 = 0.f, a3 = 0.f;
#pragma unroll
        for (int o = 0; o < ON; ++o) {
            const float* wr = Wt + o * CN + c0;   // uniform -> s_load
            const float* br = Bt + o * CN + c0;
            const float bo = bb[o];
            w0 = fmaf(bo, wr[0], w0);
            w1 = fmaf(bo, wr[1], w1);
            w2 = fmaf(bo, wr[2], w2);
            w3 = fmaf(bo, wr[3], w3);
            a0 = fmaf(bo, br[0], a0);
            a1 = fmaf(bo, br[1], a1);
            a2 = fmaf(bo, br[2], a2);
            a3 = fmaf(bo, br[3], a3);
        }
        const float wm[4] = {w0, w1, w2, w3};
        const float bm[4] = {a0, a1, a2, a3};
#pragma unroll
        for (int j = 0; j < 4; ++j) {
            const int c = c0 + j;
            const float xv = xb[(size_t)c * HW];
            const float nv = (xv - meanp[c]) * rstdp[c];
            ob[(size_t)c * HW] = fmaf(wm[j] + 1.0f, nv, bm[j]);
        }
    }
}

// ---------------------------------------------------------------------------
// Workspace layout (floats): [0,256)   mean|rstd
//                            [256, 256+10240)        weight (80x128)
//                            [10496, 10496+10240)    bias   (80x128)
// ---------------------------------------------------------------------------
extern "C" void kernel_launch(void* const* d_in, const int* in_sizes, int n_in,
                              void* d_out, int out_size, void* d_ws, size_t ws_size,
                              hipStream_t stream) {
    const float* x      = (const float*)d_in[0];
    const float* vec    = (const float*)d_in[1];
    const float* bbox   = (const float*)d_in[2];
    const float* wp_w   = (const float*)d_in[3];
    const float* wp_b   = (const float*)d_in[4];
    const float* bp_w   = (const float*)d_in[5];
    const float* bp_b   = (const float*)d_in[6];
    float* out = (float*)d_out;

    float* stats = (float*)d_ws;                 // 256 floats
    float* wmat  = stats + 2 * CN;               // 10240 floats
    float* bmat  = wmat + MROWS * CN;            // 10240 floats

    bn_stats_kernel<<<CN, 256, 0, stream>>>(x, stats);

    proj_wmma_kernel<<<dim3(CN / 16, MROWS / 16, 2), 32, 0, stream>>>(
        vec, wp_w, wp_b, bp_w, bp_b, wmat, bmat);

    fused_apply_kernel<<<dim3(HW / 256, BN), 256, 0, stream>>>(
        x, bbox, stats, wmat, bmat, out);
}